// KANCNN_60301340836310
// MI455X (gfx1250) — compile-verified
//
#include <hip/hip_runtime.h>
#include <hip/hip_bf16.h>

typedef __attribute__((ext_vector_type(2))) float v2f;
typedef __attribute__((ext_vector_type(4))) float v4f;
typedef __attribute__((ext_vector_type(8))) float v8f;

#define B_   8
#define C_   32
#define H_   64
#define W_   64
#define F_   64
#define NSL  288   // 3*3*32 (ka,kb,c) slices

// One wave32 computes one 16(spatial) x 16(filter) output tile.
// Per slice: num tile via 2x V_WMMA_F32_16X16X4_F32 (powers x^0..x^3, then
// x^4..x^5 zero-padded, chained through C), den-inner tile via 1 WMMA
// (powers x^1..x^4). VALU epilogue: acc += num * rcp(1 + |den|).
// The 9-way (ka,kb) loop is fully unrolled so clamp/bounds/address math is
// amortized over the 32-channel inner loop.
__global__ __launch_bounds__(256) void kan_rational_wmma(
    const float* __restrict__ x,
    const float* __restrict__ num_coef,   // [F,3,3,C,6] = [F, 288, 6]
    const float* __restrict__ den_coef,   // [F,3,3,C,4] = [F, 288, 4]
    float* __restrict__ out)              // [B,F,H,W]
{
    const int lane  = threadIdx.x & 31;
    const int wave  = threadIdx.x >> 5;
    const int tile  = blockIdx.x * 8 + wave;      // 8192 tiles total

    const int ftile = tile & 3;                   // 4 filter tiles of 16
    const int rtile = tile >> 2;                  // 2048 row tiles of 16
    const int base  = rtile * 16;                 // flat (b,h,w) base
    const int w0    = base & (W_ - 1);            // multiple of 16
    const int h     = (base >> 6) & (H_ - 1);
    const int b     = base >> 12;

    const int  m   = lane & 15;                   // A-row / B-col index
    const bool hiL = lane >= 16;                  // K={2,3} half per ISA layout
    const int  f   = ftile * 16 + m;              // this lane's filter column
    const int  w   = w0 + m;                      // this lane's spatial column

    // Coefficient bases with the per-half K offset folded in once.
    const int koff = hiL ? 2 : 0;
    const float* anf = num_coef + (size_t)f * (NSL * 6);
    const float* bdf = den_coef + (size_t)f * (NSL * 4);

    v8f acc = {};

#pragma unroll
    for (int kk = 0; kk < 9; ++kk) {
        const int ka = kk / 3;                    // compile-time constants
        const int kb = kk - ka * 3;

        // ---- slice-invariant bounds / clamped base address (once per kk) ----
        const int hs = h + ka - 1;
        const int ws = w + kb - 1;
        const bool inb = (hs >= 0) & (hs < H_) & (ws >= 0) & (ws < W_);
        const int hc = min(max(hs, 0), H_ - 1);
        const int wc = min(max(ws, 0), W_ - 1);
        const float* xp   = x + ((size_t)b * C_ * H_ + hc) * W_ + wc;
        const float* ap   = anf + kk * (C_ * 6) + koff;   // a_{k0},a_{k0+1}
        const float* ap45 = anf + kk * (C_ * 6) + 4;      // a4,a5
        const float* bp   = bdf + kk * (C_ * 4) + koff;   // b_{k0},b_{k0+1}

        for (int c = 0; c < C_; ++c) {
            // ---- patch value, branchless zero-pad (EXEC stays all-1s) ----
            float xv = xp[(size_t)c * (H_ * W_)];
            xv = inb ? xv : 0.0f;     // x=0 -> num=a0, den=1 (matches ref pad)

            // ---- A-matrix fragments (16x4 f32):
            //   lanes 0-15 hold K={0,1}, lanes 16-31 hold K={2,3}
            //   A1: powers 0..3   A2: powers 4,5,0,0   Ad: powers 1..4
            const float x2  = xv * xv;
            const float a1x = hiL ? x2 : 1.0f;
            v2f A1 = { a1x, a1x * xv };
            v2f Ad = { A1.x * xv, A1.y * xv };
            const float x4  = x2 * x2;
            const float a2x = hiL ? 0.0f : x4;
            v2f A2 = { a2x, a2x * xv };

            // ---- B-matrix fragments: coefficients for this filter column ----
            v2f Bn1 = *(const v2f*)(ap   + c * 6);
            v2f Bd  = *(const v2f*)(bp   + c * 4);
            v2f t45 = *(const v2f*)(ap45 + c * 6);
            v2f Bn2 = hiL ? (v2f){0.f, 0.f} : t45;

            // ---- matrix-pipe contractions ----
            v8f zero = {};
            v8f numt = __builtin_amdgcn_wmma_f32_16x16x4_f32(
                false, A1, false, Bn1, (short)0, zero, false, false);
            numt = __builtin_amdgcn_wmma_f32_16x16x4_f32(
                false, A2, false, Bn2, (short)0, numt, false, false);
            v8f dent = __builtin_amdgcn_wmma_f32_16x16x4_f32(
                false, Ad, false, Bd, (short)0, zero, false, false);

            // ---- safe-Pade epilogue: acc += num / (1 + |x*Q(x)|) ----
#pragma unroll
            for (int i = 0; i < 8; ++i) {
                const float d = 1.0f + __builtin_fabsf(dent[i]);
                acc[i] += numt[i] * __builtin_amdgcn_rcpf(d);
            }
        }
    }

    // C/D layout: VGPR i -> M = i + (lane>=16 ? 8 : 0), N = lane&15.
    // M indexes w within the tile, N indexes filter -> contiguous in w.
    float* op = out + (((size_t)b * F_ + f) * H_ + h) * W_ + w0 + (hiL ? 8 : 0);
    *(v4f*)(op)     = (v4f){acc[0], acc[1], acc[2], acc[3]};
    *(v4f*)(op + 4) = (v4f){acc[4], acc[5], acc[6], acc[7]};
}

extern "C" void kernel_launch(void* const* d_in, const int* in_sizes, int n_in,
                              void* d_out, int out_size, void* d_ws, size_t ws_size,
                              hipStream_t stream) {
    const float* x  = (const float*)d_in[0];
    const float* nc = (const float*)d_in[1];
    const float* dc = (const float*)d_in[2];
    float* out = (float*)d_out;

    // 8192 wave-tiles / 8 waves per 256-thread block = 1024 blocks
    kan_rational_wmma<<<1024, 256, 0, stream>>>(x, nc, dc, out);
}